// Attention_85366769975996
// MI455X (gfx1250) — compile-verified
//
#include <hip/hip_runtime.h>
#include <math.h>

// ---------------------------------------------------------------------------
// Seq2seq RNN + attention for MI455X (gfx1250, wave32).
//   GEMMs     -> v_wmma_f32_16x16x4_f32 (fp32 WMMA, 16x64 strip per wave)
//   GEMV dots -> v_pk_fma_f32 (VOP3P packed f32) inline asm
// ---------------------------------------------------------------------------

typedef __attribute__((ext_vector_type(2))) float v2f;
typedef __attribute__((ext_vector_type(8))) float v8f;

#define N_CLASS  32000
#define N_HIDDEN 1024
#define ENC_LEN  512
#define DEC_LEN  256

// ---------------------------------------------------------------------------
// C[M,N] = X[M,K] @ W[N,K]^T + bias[N]
// One wave -> 16 rows x 64 cols (4 WMMA accumulators, A fragment reused 4x).
// Requires M%16==0, N%64==0, K%4==0 (true for all call sites).
// ---------------------------------------------------------------------------
__global__ __launch_bounds__(256) void wmma_gemm_nt(
    const float* __restrict__ X, const float* __restrict__ W,
    const float* __restrict__ bias, float* __restrict__ C,
    int M, int N, int K)
{
    const int lane  = threadIdx.x & 31;
    const int wave  = threadIdx.x >> 5;
    const int tilesN = N >> 6;                       // 64-wide strips
    const int tilesM = M >> 4;
    const int gwave = blockIdx.x * 8 + wave;
    if (gwave >= tilesM * tilesN) return;            // wave-uniform exit
    const int tm = gwave / tilesN;
    const int tn = gwave % tilesN;

    // ISA fragment layout (32-bit, 16x4 A / 4x16 B):
    //   lanes 0-15 : row = lane,    K = k0+0 / k0+1  (VGPR0/VGPR1)
    //   lanes 16-31: row = lane-16, K = k0+2 / k0+3
    const int row = lane & 15;
    const int kp  = (lane >> 4) << 1;                // 0 or 2

    const float* xptr = X + (size_t)(tm * 16 + row) * K + kp;
    const float* wptr = W + (size_t)(tn * 64 + row) * K + kp;
    const size_t wstep = (size_t)16 * K;             // next 16-col subtile of B

    v8f c0 = {}, c1 = {}, c2 = {}, c3 = {};

#pragma unroll 2
    for (int k0 = 0; k0 < K; k0 += 4) {
        v2f a  = *(const v2f*)(xptr + k0);
        v2f b0 = *(const v2f*)(wptr + k0);
        v2f b1 = *(const v2f*)(wptr + wstep + k0);
        v2f b2 = *(const v2f*)(wptr + 2 * wstep + k0);
        v2f b3 = *(const v2f*)(wptr + 3 * wstep + k0);
        c0 = __builtin_amdgcn_wmma_f32_16x16x4_f32(false, a, false, b0, (short)0, c0, false, false);
        c1 = __builtin_amdgcn_wmma_f32_16x16x4_f32(false, a, false, b1, (short)0, c1, false, false);
        c2 = __builtin_amdgcn_wmma_f32_16x16x4_f32(false, a, false, b2, (short)0, c2, false, false);
        c3 = __builtin_amdgcn_wmma_f32_16x16x4_f32(false, a, false, b3, (short)0, c3, false, false);
    }

    // C/D layout: VGPR v -> M = v (lanes 0-15) or 8+v (lanes 16-31), N = lane&15
    const int colBase = tn * 64 + (lane & 15);
    const int rowBase = tm * 16 + ((lane >> 4) << 3);
    const float bb0 = bias[colBase];
    const float bb1 = bias[colBase + 16];
    const float bb2 = bias[colBase + 32];
    const float bb3 = bias[colBase + 48];
#pragma unroll
    for (int v = 0; v < 8; ++v) {
        float* crow = C + (size_t)(rowBase + v) * N + colBase;
        crow[0]  = c0[v] + bb0;
        crow[16] = c1[v] + bb1;
        crow[32] = c2[v] + bb2;
        crow[48] = c3[v] + bb3;
    }
}

// packed f32 dot-product helper: acc2 += a * b (per 32-bit half)
__device__ inline void pk_fma(v2f& acc, v2f a, v2f b)
{
    asm("v_pk_fma_f32 %0, %1, %2, %0" : "+v"(acc) : "v"(a), "v"(b));
}

// ---------------------------------------------------------------------------
// Encoder recurrence: h_t = tanh(xw[t] + W_hh @ h_{t-1} + b_hh), h kept in LDS.
// Single 1024-thread workgroup (one output element per thread).
// ---------------------------------------------------------------------------
__global__ __launch_bounds__(1024) void rnn_encoder(
    const float* __restrict__ xw, const float* __restrict__ Whh,
    const float* __restrict__ bhh, const float* __restrict__ h0,
    float* __restrict__ encOut, int S, int H)
{
    __shared__ float h[N_HIDDEN];
    const int j = threadIdx.x;
    h[j] = h0[j];
    __syncthreads();

    const v2f* wrow = (const v2f*)(Whh + (size_t)j * H);
    const float bj = bhh[j];

    for (int t = 0; t < S; ++t) {
        const v2f* hv = (const v2f*)h;
        v2f acc = {0.f, 0.f};
#pragma unroll 8
        for (int k = 0; k < H / 2; ++k)
            pk_fma(acc, wrow[k], hv[k]);
        float hn = tanhf(xw[(size_t)t * H + j] + bj + acc.x + acc.y);
        __syncthreads();                 // everyone done reading old h
        h[j] = hn;
        encOut[(size_t)t * H + j] = hn;
        __syncthreads();
    }
}

// ---------------------------------------------------------------------------
// Decoder recurrence + attention. Single 1024-thread workgroup.
// Per step: h update, scores = A @ h, softmax(512), context = w @ encOut,
// write [h | context] row into HC (for the batched output GEMM) and attns.
// ---------------------------------------------------------------------------
__global__ __launch_bounds__(1024) void rnn_decoder(
    const float* __restrict__ xw, const float* __restrict__ Whh,
    const float* __restrict__ bhh, const float* __restrict__ A,
    const float* __restrict__ encOut, float* __restrict__ HC,
    float* __restrict__ attns, int Sd, int Se, int H)
{
    __shared__ float h[N_HIDDEN];
    __shared__ float sc[ENC_LEN];
    __shared__ float red[ENC_LEN];
    const int j = threadIdx.x;

    h[j] = encOut[(size_t)(Se - 1) * H + j];   // h_enc == last encoder output
    __syncthreads();

    const v2f* wrow = (const v2f*)(Whh + (size_t)j * H);
    const v2f* arow = (j < Se) ? (const v2f*)(A + (size_t)j * H) : (const v2f*)A;
    const float bj = bhh[j];

    for (int t = 0; t < Sd; ++t) {
        // ---- h update ----
        const v2f* hv = (const v2f*)h;
        v2f acc = {0.f, 0.f};
#pragma unroll 8
        for (int k = 0; k < H / 2; ++k)
            pk_fma(acc, wrow[k], hv[k]);
        float hn = tanhf(xw[(size_t)t * H + j] + bj + acc.x + acc.y);
        __syncthreads();
        h[j] = hn;
        __syncthreads();

        // ---- scores = A @ h ----
        if (j < Se) {
            v2f sacc = {0.f, 0.f};
#pragma unroll 8
            for (int k = 0; k < H / 2; ++k)
                pk_fma(sacc, arow[k], hv[k]);
            float s = sacc.x + sacc.y;
            sc[j] = s;
            red[j] = s;
        }
        __syncthreads();

        // ---- softmax: max ----
        for (int off = ENC_LEN / 2; off > 0; off >>= 1) {
            if (j < off) red[j] = fmaxf(red[j], red[j + off]);
            __syncthreads();
        }
        const float mx = red[0];
        __syncthreads();

        // ---- softmax: exp + sum ----
        float e = 0.f;
        if (j < Se) { e = __expf(sc[j] - mx); red[j] = e; }
        __syncthreads();
        for (int off = ENC_LEN / 2; off > 0; off >>= 1) {
            if (j < off) red[j] += red[j + off];
            __syncthreads();
        }
        const float inv = 1.0f / red[0];
        __syncthreads();
        if (j < Se) {
            float wv = e * inv;
            sc[j] = wv;
            attns[(size_t)t * Se + j] = wv;
        }
        __syncthreads();

        // ---- context = w @ encOut (coalesced across j) ----
        float ctx = 0.f;
        for (int i = 0; i < Se; ++i)
            ctx = fmaf(sc[i], encOut[(size_t)i * H + j], ctx);

        HC[(size_t)t * 2 * H + j]     = h[j];
        HC[(size_t)t * 2 * H + H + j] = ctx;
        // next iteration's writes to h/sc are barrier-protected above
    }
}

// ---------------------------------------------------------------------------
extern "C" void kernel_launch(void* const* d_in, const int* in_sizes, int n_in,
                              void* d_out, int out_size, void* d_ws, size_t ws_size,
                              hipStream_t stream)
{
    const float* enc_input = (const float*)d_in[0];   // [512, 32000]
    const float* hidden    = (const float*)d_in[1];   // [1024]
    const float* dec_input = (const float*)d_in[2];   // [256, 32000]
    const float* enc_W_ih  = (const float*)d_in[3];   // [1024, 32000]
    const float* enc_W_hh  = (const float*)d_in[4];   // [1024, 1024]
    const float* enc_b_ih  = (const float*)d_in[5];
    const float* enc_b_hh  = (const float*)d_in[6];
    const float* dec_W_ih  = (const float*)d_in[7];   // [1024, 32000]
    const float* dec_W_hh  = (const float*)d_in[8];
    const float* dec_b_ih  = (const float*)d_in[9];
    const float* dec_b_hh  = (const float*)d_in[10];
    const float* attn_W    = (const float*)d_in[11];  // [1024, 1024]
    const float* attn_b    = (const float*)d_in[12];
    const float* out_W     = (const float*)d_in[13];  // [32000, 2048]
    const float* out_b     = (const float*)d_in[14];

    float* ws      = (float*)d_ws;
    float* xw_enc  = ws;                                   // 512*1024
    float* xw_dec  = xw_enc + (size_t)ENC_LEN * N_HIDDEN;  // 256*1024
    float* enc_out = xw_dec + (size_t)DEC_LEN * N_HIDDEN;  // 512*1024
    float* Amat    = enc_out + (size_t)ENC_LEN * N_HIDDEN; // 512*1024
    float* HC      = Amat + (size_t)ENC_LEN * N_HIDDEN;    // 256*2048

    float* outs  = (float*)d_out;                          // [256, 32000]
    float* attns = outs + (size_t)DEC_LEN * N_CLASS;       // [256, 512]

    auto gemm = [&](const float* X, const float* W, const float* b, float* C,
                    int M, int N, int K) {
        int waves  = (M >> 4) * (N >> 6);
        int blocks = (waves + 7) / 8;
        wmma_gemm_nt<<<blocks, 256, 0, stream>>>(X, W, b, C, M, N, K);
    };

    // 1) input projections (big GEMMs, K = 32000)
    gemm(enc_input, enc_W_ih, enc_b_ih, xw_enc, ENC_LEN, N_HIDDEN, N_CLASS);
    gemm(dec_input, dec_W_ih, dec_b_ih, xw_dec, DEC_LEN, N_HIDDEN, N_CLASS);

    // 2) encoder recurrence (sequential, single workgroup)
    rnn_encoder<<<1, 1024, 0, stream>>>(xw_enc, enc_W_hh, enc_b_hh, hidden,
                                        enc_out, ENC_LEN, N_HIDDEN);

    // 3) attention projection A = enc_out @ attn_W^T + attn_b
    gemm(enc_out, attn_W, attn_b, Amat, ENC_LEN, N_HIDDEN, N_HIDDEN);

    // 4) decoder recurrence + softmax attention -> HC rows + attns
    rnn_decoder<<<1, 1024, 0, stream>>>(xw_dec, dec_W_hh, dec_b_hh, Amat,
                                        enc_out, HC, attns,
                                        DEC_LEN, ENC_LEN, N_HIDDEN);

    // 5) batched output projection: outs = HC @ out_W^T + out_b
    gemm(HC, out_W, out_b, outs, DEC_LEN, N_CLASS, 2 * N_HIDDEN);
}